// HashedConv2d_60370060312715
// MI455X (gfx1250) — compile-verified
//
#include <hip/hip_runtime.h>

typedef __attribute__((ext_vector_type(2))) float v2f;
typedef __attribute__((ext_vector_type(4))) float v4f;
typedef __attribute__((ext_vector_type(8))) float v8f;

#define CIN    64
#define COUT   128
#define HW_    112
#define K_TOT  576              // 64 * 3 * 3
#define NPIX   (112 * 112)      // 12544
#define NBATCH 32
#define NGEMM  (NBATCH * NPIX)  // 401408
#define BUCKETS 16
#define NPB    32               // pixels (GEMM-N) per block: two 16-wide subtiles

// d_ws layout (in floats):
#define WS_MINMAX 0    // [0]=wmin, [1]=wmax
#define WS_WB     16   // packed weights: WB[(k>>1)*512 + co*4 + {0,1}]=W(k0,k1),
                       //                 +{2,3}=hashedW(k0,k1). 147456 floats.

// ---------------------------------------------------------------------------
// Kernel 1: min/max of W, single block (73728 elems / 1024 threads = 72 iters)
// ---------------------------------------------------------------------------
__global__ __launch_bounds__(1024) void minmax_kernel(const float* __restrict__ W,
                                                      float* __restrict__ ws) {
    __shared__ float smin[1024];
    __shared__ float smax[1024];
    int t = threadIdx.x;
    float mn = 3.4e38f, mx = -3.4e38f;
    for (int i = t; i < COUT * K_TOT; i += 1024) {
        float v = W[i];
        mn = fminf(mn, v);
        mx = fmaxf(mx, v);
    }
    smin[t] = mn;
    smax[t] = mx;
    __syncthreads();
    for (int s = 512; s > 0; s >>= 1) {
        if (t < s) {
            smin[t] = fminf(smin[t], smin[t + s]);
            smax[t] = fmaxf(smax[t], smax[t + s]);
        }
        __syncthreads();
    }
    if (t == 0) {
        ws[WS_MINMAX + 0] = smin[0];
        ws[WS_MINMAX + 1] = smax[0];
    }
}

// ---------------------------------------------------------------------------
// Kernel 2: sequential 16-bucket hashing (faithful to in-place loop semantics:
// replaced values are re-compared against later thresholds), then write BOTH
// weight sets packed per (k-pair, co) so one global_load_b128 fetches the A
// fragments of both convs:  WB[(k>>1)*4*COUT + co*4 + (k&1)]      = W
//                           WB[(k>>1)*4*COUT + co*4 + (k&1) + 2]  = hashed W
// ---------------------------------------------------------------------------
__global__ __launch_bounds__(256) void prep_kernel(const float* __restrict__ W,
                                                   const float* __restrict__ hashed,
                                                   float* __restrict__ ws) {
    int i = blockIdx.x * 256 + threadIdx.x;
    if (i >= COUT * K_TOT) return;
    float wmin = ws[WS_MINMAX + 0];
    float wmax = ws[WS_MINMAX + 1];
    float step = (wmax - wmin) * (1.0f / BUCKETS);
    float v = W[i];
    float h = v;
#pragma unroll
    for (int bkt = 0; bkt < BUCKETS; ++bkt) {
        float thr = (float)(bkt + 1) * step + wmin;
        float hv  = hashed[bkt];
        h = (h > thr) ? hv : h;
    }
    int co = i / K_TOT;
    int k  = i - co * K_TOT;
    int idx = (k >> 1) * (4 * COUT) + co * 4 + (k & 1);
    ws[WS_WB + idx]     = v;
    ws[WS_WB + idx + 2] = h;
}

// ---------------------------------------------------------------------------
// Kernel 3: implicit-GEMM conv via V_WMMA_F32_16X16X4_F32, both convs fused,
// 2 N-subtiles per wave. Block = 256 threads (8 wave32) = 32 pixels.
// LDS: full 576x32 im2col column, pair-interleaved with a parity-XOR subtile
// swizzle so the two lane-halves (rows p, p+1) hit disjoint 32-bank sets:
//     smem[(k>>1)*64 + (((k>>1)&1)^nt)*32 + n*2 + (k&1)]
// Steady state per K+=4: 1 b128 global + 2 b64 LDS + 4 WMMA.
// All per-lane access streams are affine in k0 -> expressed as running
// pointers with constant strides so addressing folds into immediates.
// ---------------------------------------------------------------------------
__global__ __launch_bounds__(256) void conv_kernel(const float* __restrict__ x,
                                                   const float* __restrict__ b,
                                                   const float* __restrict__ b2,
                                                   const float* __restrict__ ws,
                                                   float* __restrict__ out) {
    __shared__ float smem[(K_TOT / 2) * 64];   // 18432 floats = 72 KB

    const float* WB = ws + WS_WB;

    const int t    = threadIdx.x;
    const int lane = t & 31;
    const int wave = t >> 5;
    const int n_l  = lane & 15;
    const int half = lane >> 4;

    const int n_base = blockIdx.x * NPB;

    // ---- cooperative im2col: thread t -> pixel (t&31), k = (t>>5) + 8j
    {
        int ln = t & 31;
        int kb = t >> 5;                  // 0..7
        int nt = ln >> 4;                 // subtile
        int nn = ln & 15;
        int ng = n_base + ln;
        int nb = ng / NPIX;
        int rem = ng - nb * NPIX;
        int oh = rem / HW_;
        int ow = rem - oh * HW_;
        const float* xb = x + (size_t)nb * CIN * NPIX;
#pragma unroll 4
        for (int j = 0; j < K_TOT / 8; ++j) {
            int k  = kb + j * 8;
            int ci = k / 9;
            int r9 = k - ci * 9;
            int dy = r9 / 3;
            int dx = r9 - dy * 3;
            int ih = oh + dy - 1;
            int iw = ow + dx - 1;
            float v = 0.0f;
            if (ih >= 0 && ih < HW_ && iw >= 0 && iw < HW_)
                v = xb[(ci * HW_ + ih) * HW_ + iw];
            int p = k >> 1;
            smem[p * 64 + (((p & 1) ^ nt) << 5) + nn * 2 + (k & 1)] = v;
        }
    }
    __syncthreads();

    // ---- WMMA main loop (running pointers; strides constant per step)
    const int co_base = wave * 16;
    const int sw      = half << 5;   // (p&1)*32, loop-invariant since p&1==half

    const float* pa  = WB + (size_t)half * (4 * COUT) + (co_base + n_l) * 4;
    const float* pb0 = smem + (size_t)half * 64 + sw        + n_l * 2;
    const float* pb1 = smem + (size_t)half * 64 + (sw ^ 32) + n_l * 2;

    v8f c00 = {}, c01 = {}, c10 = {}, c11 = {};
#pragma unroll 4
    for (int k0 = 0; k0 < K_TOT; k0 += 4) {
        v4f ab = *(const v4f*)pa;    // W k-pair (.xy) + hashed-W k-pair (.zw)
        v2f b0 = *(const v2f*)pb0;   // B fragment, subtile 0
        v2f b1 = *(const v2f*)pb1;   // B fragment, subtile 1
        v2f a0 = ab.xy;
        v2f a1 = ab.zw;
        c00 = __builtin_amdgcn_wmma_f32_16x16x4_f32(false, a0, false, b0,
                                                    (short)0, c00, false, false);
        c01 = __builtin_amdgcn_wmma_f32_16x16x4_f32(false, a0, false, b1,
                                                    (short)0, c01, false, false);
        c10 = __builtin_amdgcn_wmma_f32_16x16x4_f32(false, a1, false, b0,
                                                    (short)0, c10, false, false);
        c11 = __builtin_amdgcn_wmma_f32_16x16x4_f32(false, a1, false, b1,
                                                    (short)0, c11, false, false);
        pa  += 2 * (4 * COUT);   // advance 2 k-pairs
        pb0 += 128;
        pb1 += 128;
    }

    // ---- epilogue: bias + store. C/D layout: vgpr r -> M = r + 8*half
    float* out0 = out;
    float* out1 = out + (size_t)NBATCH * COUT * NPIX;
#pragma unroll
    for (int nt = 0; nt < 2; ++nt) {
        int ng = n_base + nt * 16 + n_l;
        int nb = ng / NPIX;
        int rem = ng - nb * NPIX;
        v8f d0 = nt ? c01 : c00;
        v8f d1 = nt ? c11 : c10;
#pragma unroll
        for (int r = 0; r < 8; ++r) {
            int co = co_base + r + 8 * half;
            size_t o = (size_t)(nb * COUT + co) * NPIX + (size_t)rem;
            out0[o] = d0[r] + b[co];
            out1[o] = d1[r] + b2[co];
        }
    }
}

// ---------------------------------------------------------------------------
extern "C" void kernel_launch(void* const* d_in, const int* in_sizes, int n_in,
                              void* d_out, int out_size, void* d_ws, size_t ws_size,
                              hipStream_t stream) {
    const float* x      = (const float*)d_in[0];
    const float* W      = (const float*)d_in[1];
    const float* b      = (const float*)d_in[2];
    const float* b2     = (const float*)d_in[3];
    const float* hashed = (const float*)d_in[4];
    float* ws  = (float*)d_ws;
    float* out = (float*)d_out;

    minmax_kernel<<<1, 1024, 0, stream>>>(W, ws);
    prep_kernel<<<(COUT * K_TOT + 255) / 256, 256, 0, stream>>>(W, hashed, ws);
    conv_kernel<<<NGEMM / NPB, 256, 0, stream>>>(x, b, b2, ws, out);
}